// S5Layer_85220741087274
// MI455X (gfx1250) — compile-verified
//
#include <hip/hip_runtime.h>
#include <cstddef>

// ---- problem constants (match reference) ----
#define PDIM   256
#define HDIM   256
#define BSZ    16
#define LSEQ   4096
#define BL     (BSZ * LSEQ)        // 65536 rows
#define NCH    16                  // scan chunks per sequence
#define CLEN   (LSEQ / NCH)        // 256 steps per chunk
#define PHN    (PDIM * HDIM)       // 65536
#define BNP    (BSZ * NCH * PDIM)  // 65536

typedef __attribute__((ext_vector_type(2))) float v2f;
typedef __attribute__((ext_vector_type(8))) float v8f;

#define WMMA_F32(a, b, c) \
    __builtin_amdgcn_wmma_f32_16x16x4_f32(false, (a), false, (b), (short)0, (c), false, false)

// =====================================================================
// Kernel 0: per-p discretization params.
//   LB = exp(Lam*step); AP = LB^CLEN (analytic inter-chunk factor);
//   CF = (LB - 1)/Lam  (complex)
// =====================================================================
__global__ void k_params(const float* __restrict__ Lre, const float* __restrict__ Lim,
                         const float* __restrict__ logstep,
                         float* __restrict__ LBre, float* __restrict__ LBim,
                         float* __restrict__ APre, float* __restrict__ APim,
                         float* __restrict__ CFre, float* __restrict__ CFim) {
    int p = threadIdx.x;
    float st = __expf(logstep[p]);            // STEP_RESCALE == 1
    float lr = Lre[p], li = Lim[p];
    float er = lr * st, ei = li * st;         // Lam*step
    float mag = __expf(er);
    float lbr = mag * __cosf(ei);
    float lbi = mag * __sinf(ei);
    LBre[p] = lbr; LBim[p] = lbi;
    float magN = __expf(er * (float)CLEN);    // exp(CLEN*Lam*step)
    APre[p] = magN * __cosf(ei * (float)CLEN);
    APim[p] = magN * __sinf(ei * (float)CLEN);
    float nr = lbr - 1.0f, ni = lbi;          // (LB - 1)
    float den = lr * lr + li * li;
    CFre[p] = (nr * lr + ni * li) / den;      // * conj(Lam)/|Lam|^2
    CFim[p] = (ni * lr - nr * li) / den;
}

// =====================================================================
// Kernel 1: build PAIR-PACKED WMMA operand matrices so each B-fragment
// (rows k,k+1 at column n) is one aligned b64 load:
//   element (k, n) of an (K x N) operand lives at ((k>>1)*N + n)*2 + (k&1)
//   BT (K=H, N=P):  BT(h,p) = CF[p] * (B_re + i B_im)[p,h]
//   CT (K=P, N=H):  CT(p,h) = C[h,p]
// =====================================================================
__global__ void k_mats(const float* __restrict__ Bre, const float* __restrict__ Bim,
                       const float* __restrict__ Cre, const float* __restrict__ Cim,
                       const float* __restrict__ CFre, const float* __restrict__ CFim,
                       float* __restrict__ BTre, float* __restrict__ BTim,
                       float* __restrict__ CTre, float* __restrict__ CTim) {
    int i = blockIdx.x * 256 + threadIdx.x;   // 0..PHN-1
    {   // BT: k = h (0..255), n = p
        int p = i & 255, h = i >> 8;
        float cr = CFre[p], ci = CFim[p];
        float br = Bre[p * HDIM + h], bi = Bim[p * HDIM + h];
        size_t o = (size_t)((h >> 1) * PDIM + p) * 2 + (h & 1);
        BTre[o] = cr * br - ci * bi;
        BTim[o] = cr * bi + ci * br;
    }
    {   // CT: k = p, n = h
        int p = i >> 8, h = i & 255;
        size_t o = (size_t)((p >> 1) * HDIM + h) * 2 + (p & 1);
        CTre[o] = Cre[h * PDIM + p];
        CTim[o] = Cim[h * PDIM + p];
    }
}

// =====================================================================
// Kernel 2: GEMM #1 via V_WMMA_F32_16X16X4_F32.
//   Bu_{re,im}[m,p] = sum_h u[m,h] * BT_{re,im}(h,p)
// One wave -> 32x32 tile (2x2 sub-tiles, re+im): 8 accumulators,
// 6 b64 loads per k-step feeding 8 WMMAs.
// =====================================================================
__global__ void __launch_bounds__(256)
k_gemm1(const float* __restrict__ u,
        const float* __restrict__ BTre, const float* __restrict__ BTim,
        float* __restrict__ BUre, float* __restrict__ BUim) {
    int lane = threadIdx.x & 31, wid = threadIdx.x >> 5;
    int tile = blockIdx.x * 8 + wid;          // (BL/32)*(P/32) = 16384 tiles
    int m0 = (tile >> 3) << 5;                // 32-row tile
    int n0 = (tile & 7) << 5;                 // 32-col tile
    int half = lane >> 4, l16 = lane & 15;
    const float* A0 = u + (size_t)(m0 + l16) * HDIM;
    const float* A1 = A0 + (size_t)16 * HDIM;
    int n = n0 + l16;

    v8f accR[2][2] = {{{}, {}}, {{}, {}}};
    v8f accI[2][2] = {{{}, {}}, {{}, {}}};
#pragma unroll 4
    for (int k0 = 0; k0 < HDIM; k0 += 4) {
        int k  = k0 + 2 * half;               // even
        size_t kp = (size_t)(k >> 1) * PDIM;  // pair-packed row base
        v2f a0  = *(const v2f*)(A0 + k);
        v2f a1  = *(const v2f*)(A1 + k);
        v2f br0 = *(const v2f*)(BTre + (kp + n) * 2);
        v2f br1 = *(const v2f*)(BTre + (kp + n + 16) * 2);
        v2f bi0 = *(const v2f*)(BTim + (kp + n) * 2);
        v2f bi1 = *(const v2f*)(BTim + (kp + n + 16) * 2);
        accR[0][0] = WMMA_F32(a0, br0, accR[0][0]);
        accR[0][1] = WMMA_F32(a0, br1, accR[0][1]);
        accR[1][0] = WMMA_F32(a1, br0, accR[1][0]);
        accR[1][1] = WMMA_F32(a1, br1, accR[1][1]);
        accI[0][0] = WMMA_F32(a0, bi0, accI[0][0]);
        accI[0][1] = WMMA_F32(a0, bi1, accI[0][1]);
        accI[1][0] = WMMA_F32(a1, bi0, accI[1][0]);
        accI[1][1] = WMMA_F32(a1, bi1, accI[1][1]);
    }
#pragma unroll
    for (int mi = 0; mi < 2; ++mi)
#pragma unroll
        for (int j = 0; j < 8; ++j) {
            int m = m0 + mi * 16 + j + 8 * half;
            size_t o = (size_t)m * PDIM + n;
            BUre[o]      = accR[mi][0][j];
            BUre[o + 16] = accR[mi][1][j];
            BUim[o]      = accI[mi][0][j];
            BUim[o + 16] = accI[mi][1][j];
        }
}

// =====================================================================
// Kernel 3: chunk-local inclusive scans, in place over Bu.
//   x_t = LB * x_{t-1} + Bu_t, x_{-1}=0 (per chunk). Save chunk-last.
// =====================================================================
__global__ void k_scan_local(float* __restrict__ BUre, float* __restrict__ BUim,
                             const float* __restrict__ LBre, const float* __restrict__ LBim,
                             float* __restrict__ LSre, float* __restrict__ LSim) {
    int p = threadIdx.x;
    int b = blockIdx.x >> 4, ch = blockIdx.x & 15;
    float ar = LBre[p], ai = LBim[p];
    float xr = 0.f, xi = 0.f;
    size_t base = (size_t)(b * LSEQ + ch * CLEN) * PDIM + p;
    for (int t = 0; t < CLEN; ++t) {
        size_t o = base + (size_t)t * PDIM;
        float br = BUre[o], bi = BUim[o];
        float nxr = ar * xr - ai * xi + br;
        float nxi = ar * xi + ai * xr + bi;
        xr = nxr; xi = nxi;
        BUre[o] = xr; BUim[o] = xi;
    }
    int ci = (b * NCH + ch) * PDIM + p;
    LSre[ci] = xr; LSim[ci] = xi;
}

// =====================================================================
// Kernel 4: sequential carry combine across the 16 chunks (per b,p).
//   carry_in[ch] = AP * carry_in[ch-1] + last[ch-1]
// =====================================================================
__global__ void k_carry(const float* __restrict__ LSre, const float* __restrict__ LSim,
                        const float* __restrict__ APre, const float* __restrict__ APim,
                        float* __restrict__ CIre, float* __restrict__ CIim) {
    int p = threadIdx.x, b = blockIdx.x;
    float apr = APre[p], api = APim[p];
    float cr = 0.f, ci = 0.f;
    for (int ch = 0; ch < NCH; ++ch) {
        int idx = (b * NCH + ch) * PDIM + p;
        CIre[idx] = cr; CIim[idx] = ci;
        float lr = LSre[idx], li = LSim[idx];
        float nr = apr * cr - api * ci + lr;
        float ni = apr * ci + api * cr + li;
        cr = nr; ci = ni;
    }
}

// =====================================================================
// Kernel 5: fix-up: xs_t += LB^{t_local+1} * carry_in[chunk]
// =====================================================================
__global__ void k_fixup(float* __restrict__ BUre, float* __restrict__ BUim,
                        const float* __restrict__ LBre, const float* __restrict__ LBim,
                        const float* __restrict__ CIre, const float* __restrict__ CIim) {
    int p = threadIdx.x;
    int b = blockIdx.x >> 4, ch = blockIdx.x & 15;
    if (ch == 0) return;                      // carry is exactly zero
    int ci = (b * NCH + ch) * PDIM + p;
    float cr = CIre[ci], cim = CIim[ci];
    float lbr = LBre[p], lbi = LBim[p];
    float ar = lbr, ai = lbi;                 // LB^1
    size_t base = (size_t)(b * LSEQ + ch * CLEN) * PDIM + p;
    for (int t = 0; t < CLEN; ++t) {
        size_t o = base + (size_t)t * PDIM;
        BUre[o] += ar * cr - ai * cim;
        BUim[o] += ar * cim + ai * cr;
        float nar = ar * lbr - ai * lbi;      // a *= LB
        float nai = ar * lbi + ai * lbr;
        ar = nar; ai = nai;
    }
}

// =====================================================================
// Kernel 6: GEMM #2 via WMMA + epilogue.
//   y[m,h] = 2*( xs_re@CT_re - xs_im@CT_im )[m,h] + D[h]*u[m,h]
// 32x32 per-wave tile; xs rows give contiguous (k,k+1) pairs (lda=P),
// CT is pair-packed -> all fragment loads are b64.
// =====================================================================
__global__ void __launch_bounds__(256)
k_gemm2(const float* __restrict__ XSre, const float* __restrict__ XSim,
        const float* __restrict__ CTre, const float* __restrict__ CTim,
        const float* __restrict__ D, const float* __restrict__ u,
        float* __restrict__ y) {
    int lane = threadIdx.x & 31, wid = threadIdx.x >> 5;
    int tile = blockIdx.x * 8 + wid;          // (BL/32)*(H/32) = 16384 tiles
    int m0 = (tile >> 3) << 5;
    int n0 = (tile & 7) << 5;
    int half = lane >> 4, l16 = lane & 15;
    const float* AR0 = XSre + (size_t)(m0 + l16) * PDIM;
    const float* AR1 = AR0 + (size_t)16 * PDIM;
    const float* AI0 = XSim + (size_t)(m0 + l16) * PDIM;
    const float* AI1 = AI0 + (size_t)16 * PDIM;
    int n = n0 + l16;

    v8f accR[2][2] = {{{}, {}}, {{}, {}}};
    v8f accI[2][2] = {{{}, {}}, {{}, {}}};
#pragma unroll 4
    for (int k0 = 0; k0 < PDIM; k0 += 4) {
        int k  = k0 + 2 * half;
        size_t kp = (size_t)(k >> 1) * HDIM;
        v2f ar0 = *(const v2f*)(AR0 + k);
        v2f ar1 = *(const v2f*)(AR1 + k);
        v2f ai0 = *(const v2f*)(AI0 + k);
        v2f ai1 = *(const v2f*)(AI1 + k);
        v2f br0 = *(const v2f*)(CTre + (kp + n) * 2);
        v2f br1 = *(const v2f*)(CTre + (kp + n + 16) * 2);
        v2f bi0 = *(const v2f*)(CTim + (kp + n) * 2);
        v2f bi1 = *(const v2f*)(CTim + (kp + n + 16) * 2);
        accR[0][0] = WMMA_F32(ar0, br0, accR[0][0]);
        accR[0][1] = WMMA_F32(ar0, br1, accR[0][1]);
        accR[1][0] = WMMA_F32(ar1, br0, accR[1][0]);
        accR[1][1] = WMMA_F32(ar1, br1, accR[1][1]);
        accI[0][0] = WMMA_F32(ai0, bi0, accI[0][0]);
        accI[0][1] = WMMA_F32(ai0, bi1, accI[0][1]);
        accI[1][0] = WMMA_F32(ai1, bi0, accI[1][0]);
        accI[1][1] = WMMA_F32(ai1, bi1, accI[1][1]);
    }
    float d0 = D[n], d1 = D[n + 16];
#pragma unroll
    for (int mi = 0; mi < 2; ++mi)
#pragma unroll
        for (int j = 0; j < 8; ++j) {
            int m = m0 + mi * 16 + j + 8 * half;
            size_t o = (size_t)m * HDIM + n;
            y[o]      = 2.0f * (accR[mi][0][j] - accI[mi][0][j]) + d0 * u[o];
            y[o + 16] = 2.0f * (accR[mi][1][j] - accI[mi][1][j]) + d1 * u[o + 16];
        }
}

// =====================================================================
// Launch: params -> operand prep -> GEMM1 -> local scan -> carries ->
//         fix-up -> GEMM2.  Workspace needs ~136.5 MB.
// =====================================================================
extern "C" void kernel_launch(void* const* d_in, const int* in_sizes, int n_in,
                              void* d_out, int out_size, void* d_ws, size_t ws_size,
                              hipStream_t stream) {
    const float* u    = (const float*)d_in[0];
    const float* Lre  = (const float*)d_in[1];
    const float* Lim  = (const float*)d_in[2];
    const float* Bre  = (const float*)d_in[3];
    const float* Bim  = (const float*)d_in[4];
    const float* Cre  = (const float*)d_in[5];
    const float* Cim  = (const float*)d_in[6];
    const float* Dv   = (const float*)d_in[7];
    const float* lstp = (const float*)d_in[8];
    float* y = (float*)d_out;

    // ---- workspace partition (floats) ----
    float* ws   = (float*)d_ws;
    size_t blp  = (size_t)BL * PDIM;          // 16,777,216 floats
    float* BUre = ws;                float* BUim = BUre + blp;
    float* BTre = BUim + blp;        float* BTim = BTre + PHN;
    float* CTre = BTim + PHN;        float* CTim = CTre + PHN;
    float* LBre = CTim + PHN;        float* LBim = LBre + PDIM;
    float* APre = LBim + PDIM;       float* APim = APre + PDIM;
    float* CFre = APim + PDIM;       float* CFim = CFre + PDIM;
    float* LSre = CFim + PDIM;       float* LSim = LSre + BNP;
    float* CIre = LSim + BNP;        float* CIim = CIre + BNP;

    k_params<<<1, 256, 0, stream>>>(Lre, Lim, lstp, LBre, LBim, APre, APim, CFre, CFim);
    k_mats<<<PHN / 256, 256, 0, stream>>>(Bre, Bim, Cre, Cim, CFre, CFim, BTre, BTim, CTre, CTim);
    k_gemm1<<<(BL / 32) * (PDIM / 32) / 8, 256, 0, stream>>>(u, BTre, BTim, BUre, BUim);
    k_scan_local<<<BSZ * NCH, 256, 0, stream>>>(BUre, BUim, LBre, LBim, LSre, LSim);
    k_carry<<<BSZ, 256, 0, stream>>>(LSre, LSim, APre, APim, CIre, CIim);
    k_fixup<<<BSZ * NCH, 256, 0, stream>>>(BUre, BUim, LBre, LBim, CIre, CIim);
    k_gemm2<<<(BL / 32) * (HDIM / 32) / 8, 256, 0, stream>>>(BUre, BUim, CTre, CTim, Dv, u, y);
}